// MolecularAnalyser_79706003079224
// MI455X (gfx1250) — compile-verified
//
#include <hip/hip_runtime.h>
#include <hip/hip_bf16.h>
#include <math.h>

#define NB 8
#define NZ 4
#define NX 32
#define NY 32
#define NN 4096          // NZ*NX*NY
#define NW 128           // NN/32 bitmask words

typedef float v2f __attribute__((ext_vector_type(2)));
typedef float v8f __attribute__((ext_vector_type(8)));

// ---------------------------------------------------------------------------
// K1: field extraction for pred and targ, sort keys, init first_tg
// ---------------------------------------------------------------------------
__global__ void k_prep(const float* __restrict__ pred, const float* __restrict__ targ,
                       float* __restrict__ key, float* __restrict__ posu,
                       float* __restrict__ Ru, float* __restrict__ tposr,
                       int* __restrict__ tact, float* __restrict__ tR,
                       float* __restrict__ tn2, int* __restrict__ ftg) {
  int gid = blockIdx.x * blockDim.x + threadIdx.x;      // b*NN + n
  if (gid >= NB * NN) return;
  int n = gid & (NN - 1);
  int z = n >> 10, x = (n >> 5) & 31, y = n & 31;
  const float* p = pred + (size_t)gid * 10;
  const float* tg = targ + (size_t)gid * 10;

  // prediction: sigmoid conf, normalized position, rotation rows
  float conf = 1.0f / (1.0f + expf(-p[0]));
  key[gid] = (conf > 0.5f) ? conf : -__builtin_inff();
  float px = (p[1] + (float)z) * 0.25f;
  float py = (p[2] + (float)x) * 0.03125f;
  float pz = (p[3] + (float)y) * 0.03125f;
  posu[(size_t)gid*3+0] = px; posu[(size_t)gid*3+1] = py; posu[(size_t)gid*3+2] = pz;
  float ax=p[4],ay=p[5],az=p[6], bx=p[7],by=p[8],bz=p[9];
  float cx = ay*bz - az*by, cy = az*bx - ax*bz, cz = ax*by - ay*bx;
  float* R = Ru + (size_t)gid*9;
  R[0]=ax; R[1]=ay; R[2]=az; R[3]=bx; R[4]=by; R[5]=bz; R[6]=cx; R[7]=cy; R[8]=cz;

  // target: raw conf, real-coordinate position, rotation rows, norm
  tact[gid] = (tg[0] > 0.5f) ? 1 : 0;
  float tx = (tg[1] + (float)z) * 0.25f    * 25.0f;
  float ty = (tg[2] + (float)x) * 0.03125f * 25.0f;
  float tz = (tg[3] + (float)y) * 0.03125f * 4.0f;
  tposr[(size_t)gid*3+0]=tx; tposr[(size_t)gid*3+1]=ty; tposr[(size_t)gid*3+2]=tz;
  tn2[gid] = tx*tx + ty*ty + tz*tz;
  float tax=tg[4],tay=tg[5],taz=tg[6], tbx=tg[7],tby=tg[8],tbz=tg[9];
  float tcx = tay*tbz - taz*tby, tcy = taz*tbx - tax*tbz, tcz = tax*tby - tay*tbx;
  float* TR = tR + (size_t)gid*9;
  TR[0]=tax; TR[1]=tay; TR[2]=taz; TR[3]=tbx; TR[4]=tby; TR[5]=tbz; TR[6]=tcx; TR[7]=tcy; TR[8]=tcz;

  ftg[gid] = NN;   // "no match yet"
}

// ---------------------------------------------------------------------------
// K2: per-batch bitonic argsort (descending by key) + gather sorted pred data
// ---------------------------------------------------------------------------
__global__ void k_sortgather(const float* __restrict__ key,
                             const float* __restrict__ posu, const float* __restrict__ Ru,
                             float* __restrict__ poss, float* __restrict__ posr,
                             float* __restrict__ Rs, float* __restrict__ pn2n,
                             float* __restrict__ pn2r, unsigned* __restrict__ keepb) {
  __shared__ float sk[NN];
  __shared__ int   si[NN];
  const int NT = 1024;
  int b = blockIdx.x, t = threadIdx.x;
  size_t base = (size_t)b * NN;
  for (int i = t; i < NN; i += NT) { sk[i] = key[base + i]; si[i] = i; }
  __syncthreads();
  for (int k = 2; k <= NN; k <<= 1) {
    for (int j = k >> 1; j > 0; j >>= 1) {
      for (int i = t; i < NN; i += NT) {
        int pp = i ^ j;
        if (pp > i) {
          bool up = ((i & k) == 0);
          float k0 = sk[i], k1 = sk[pp];
          bool sw = up ? (k0 < k1) : (k0 > k1);   // overall descending
          if (sw) {
            sk[i] = k1; sk[pp] = k0;
            int tmp = si[i]; si[i] = si[pp]; si[pp] = tmp;
          }
        }
      }
      __syncthreads();
    }
  }
  // initial active bitmask a0 (sorted order): active <=> key finite
  for (int w = t; w < NW; w += NT) {
    unsigned bits = 0;
    for (int l = 0; l < 32; ++l)
      if (sk[w * 32 + l] > -1e37f) bits |= (1u << l);
    keepb[b * NW + w] = bits;
  }
  // gather sorted pred fields + norms (normalized and real coords)
  for (int i = t; i < NN; i += NT) {
    int o = si[i];
    float p0 = posu[(base + o)*3+0];
    float p1 = posu[(base + o)*3+1];
    float p2 = posu[(base + o)*3+2];
    poss[(base + i)*3+0] = p0; poss[(base + i)*3+1] = p1; poss[(base + i)*3+2] = p2;
    float r0 = p0 * 25.0f, r1 = p1 * 25.0f, r2 = p2 * 4.0f;
    posr[(base + i)*3+0] = r0; posr[(base + i)*3+1] = r1; posr[(base + i)*3+2] = r2;
    pn2n[base + i] = p0*p0 + p1*p1 + p2*p2;
    pn2r[base + i] = r0*r0 + r1*r1 + r2*r2;
    for (int q = 0; q < 9; ++q) Rs[(base + i)*9 + q] = Ru[(base + o)*9 + q];
  }
}

// ---------------------------------------------------------------------------
// K3a: zero the per-batch transposed adjacency bitmask
// ---------------------------------------------------------------------------
__global__ void k_zero(unsigned* __restrict__ adjT) {
  size_t i = (size_t)blockIdx.x * blockDim.x + threadIdx.x;
  if (i < (size_t)NN * NW) adjT[i] = 0u;
}

// ---------------------------------------------------------------------------
// K3b: WMMA adjacency build.  One wave = one 16x16 tile of the Gram matrix
// G = P * P^T (normalized coords) via V_WMMA_F32_16X16X4_F32 (K=4, pad=0).
// adjT[j] holds bits over i with (||pi-pj|| < 2) && (i < j).
// A layout: lane m=lane&15, vgpr{K0,K1} (lanes 0-15) / {K2,K3} (lanes 16-31).
// B mirrors A with N=lane&15.  D: lane owns column n, rows v + (hi?8:0).
// ---------------------------------------------------------------------------
__global__ void k_adj(const float* __restrict__ poss_b, const float* __restrict__ pn2_b,
                      unsigned* __restrict__ adjT) {
  int wave = threadIdx.x >> 5, lane = threadIdx.x & 31;
  int tileI = blockIdx.x * 8 + wave;   // row tile (i)
  int tileJ = blockIdx.y;              // col tile (j)
  if (tileI > tileJ) return;           // strictly upper triangular graph
  int ibase = tileI * 16, jbase = tileJ * 16;
  int m = lane & 15; bool hi = lane >= 16;
  const float* q  = poss_b + (size_t)(ibase + m) * 3;
  const float* pp = poss_b + (size_t)(jbase + m) * 3;
  v2f a, bb;
  a[0]  = hi ? q[2]  : q[0];  a[1]  = hi ? 0.0f : q[1];
  bb[0] = hi ? pp[2] : pp[0]; bb[1] = hi ? 0.0f : pp[1];
  v8f c = {};
  c = __builtin_amdgcn_wmma_f32_16x16x4_f32(false, a, false, bb, (short)0, c, false, false);
  int j = jbase + m;
  float npj = pn2_b[j];
  int rb = ibase + (hi ? 8 : 0);
  unsigned bits = 0;
#pragma unroll
  for (int v = 0; v < 8; ++v) {
    int i = rb + v;
    float d2 = pn2_b[i] + npj - 2.0f * c[v];
    if ((d2 < 4.0f) && (i < j)) bits |= (1u << v);
  }
  if (bits) {
    int shift = (ibase & 31) + (hi ? 8 : 0);
    atomicOr(&adjT[(size_t)j * NW + (ibase >> 5)], bits << shift);
  }
}

// ---------------------------------------------------------------------------
// K4: per-batch NMS fixed point on bitmasks (one workgroup).
// supp[i] = a[i] && exists k: adj[k][i] & a[k]            (r[i] != 0)
// a'[j]   = a[j] && !exists i: adj[i][j] & a[i] & !supp[i] ((r-s)==0)
// loop until popcount(a) stops changing (matches the lax.while_loop).
// ---------------------------------------------------------------------------
__global__ void k_nms(const unsigned* __restrict__ adjT, unsigned* __restrict__ keepb_b) {
  __shared__ unsigned aw[NW], sw[NW], nw[NW];
  __shared__ int cnt;
  int t = threadIdx.x;
  if (t < NW) aw[t] = keepb_b[t];
  if (t == 0) cnt = 0;
  __syncthreads();
  if (t < NW) atomicAdd(&cnt, __popc(aw[t]));
  __syncthreads();
  int prev = cnt;
  __syncthreads();
  while (true) {
    if (t < NW) { sw[t] = 0; nw[t] = 0; }
    if (t == 0) cnt = 0;
    __syncthreads();
    // pass 1: suppressed-suppressor mask
    for (int i = t; i < NN; i += 1024) {
      if ((aw[i >> 5] >> (i & 31)) & 1u) {
        const unsigned* row = adjT + (size_t)i * NW;
        bool any = false;
        for (int w = 0; w < NW; ++w) if (row[w] & aw[w]) { any = true; break; }
        if (any) atomicOr(&sw[i >> 5], 1u << (i & 31));
      }
    }
    __syncthreads();
    // pass 2: survive iff no active, unsuppressed higher-conf neighbor
    for (int j = t; j < NN; j += 1024) {
      if ((aw[j >> 5] >> (j & 31)) & 1u) {
        const unsigned* row = adjT + (size_t)j * NW;
        bool bad = false;
        for (int w = 0; w < NW; ++w) if (row[w] & aw[w] & ~sw[w]) { bad = true; break; }
        if (!bad) atomicOr(&nw[j >> 5], 1u << (j & 31));
      }
    }
    __syncthreads();
    if (t < NW) { aw[t] = nw[t]; atomicAdd(&cnt, __popc(nw[t])); }
    __syncthreads();
    int c = cnt;
    __syncthreads();
    if (c == prev) break;   // uniform across block
    prev = c;
  }
  if (t < NW) keepb_b[t] = aw[t];
}

// ---------------------------------------------------------------------------
// K5: WMMA matching.  Tiles of tg_real x pd_real^T dot products; a pred p is
// matched by target t iff dist<2 && tg_active[t] && keep[p]; first_tg[p] is
// the minimum matching t (== argmax(within, 0)).
// ---------------------------------------------------------------------------
__global__ void k_match(const float* __restrict__ tposr, const float* __restrict__ tn2,
                        const int* __restrict__ tact, const float* __restrict__ posr,
                        const float* __restrict__ pn2r, const unsigned* __restrict__ keepb,
                        int* __restrict__ ftg) {
  int wave = threadIdx.x >> 5, lane = threadIdx.x & 31;
  int tileT = blockIdx.x * 8 + wave;
  int tileP = blockIdx.y;
  int b = blockIdx.z;
  size_t base = (size_t)b * NN;
  int tb = tileT * 16, pb = tileP * 16;
  int m = lane & 15; bool hi = lane >= 16;
  const float* q  = tposr + (base + tb + m) * 3;
  const float* pp = posr  + (base + pb + m) * 3;
  v2f a, bb;
  a[0]  = hi ? q[2]  : q[0];  a[1]  = hi ? 0.0f : q[1];
  bb[0] = hi ? pp[2] : pp[0]; bb[1] = hi ? 0.0f : pp[1];
  v8f c = {};
  c = __builtin_amdgcn_wmma_f32_16x16x4_f32(false, a, false, bb, (short)0, c, false, false);
  int p = pb + m;
  bool keep = (keepb[b * NW + (p >> 5)] >> (p & 31)) & 1u;
  if (!keep) return;                 // divergence only after the wave-wide WMMA
  float np = pn2r[base + p];
  int rb = tb + (hi ? 8 : 0);
  int mint = NN;
#pragma unroll
  for (int v = 0; v < 8; ++v) {
    int tt = rb + v;
    float d2 = tn2[base + tt] + np - 2.0f * c[v];
    if (d2 < 4.0f && tact[base + tt]) mint = min(mint, tt);
  }
  if (mint < NN) atomicMin(&ftg[base + p], mint);
}

// ---------------------------------------------------------------------------
// K6: per-batch confusion matrix + mean rotation-angle error (deterministic
// tree reduction).  out[0..23] = cm, out[24..31] = ms.
// ---------------------------------------------------------------------------
__global__ void k_final(const int* __restrict__ tact, const unsigned* __restrict__ keepb,
                        const int* __restrict__ ftg, const float* __restrict__ Rs,
                        const float* __restrict__ tR, float* __restrict__ out) {
  __shared__ int   s_ntg[256];
  __shared__ int   s_tp[256];
  __shared__ float s_ang[256];
  int b = blockIdx.x, t = threadIdx.x;
  size_t base = (size_t)b * NN;
  int ntg = 0, tp = 0; float ang = 0.0f;
  for (int n = t; n < NN; n += 256) {
    ntg += tact[base + n];
    int ft = ftg[base + n];
    if (ft < NN) {
      tp += 1;
      const float* pr = Rs + (base + n) * 9;
      const float* tr = tR + (base + ft) * 9;
#pragma unroll
      for (int r = 0; r < 3; ++r) {
        float ax = pr[r*3+0], ay = pr[r*3+1], az = pr[r*3+2];
        float bx = tr[r*3+0], by = tr[r*3+1], bz = tr[r*3+2];
        float dot = ax*bx + ay*by + az*bz;
        float na = sqrtf(ax*ax + ay*ay + az*az);
        float nb = sqrtf(bx*bx + by*by + bz*bz);
        float cosv = fminf(1.0f, fmaxf(-1.0f, dot / (na * nb)));
        ang += acosf(cosv) * (180.0f / 3.14159265358979323846f);
      }
    }
  }
  s_ntg[t] = ntg; s_tp[t] = tp; s_ang[t] = ang;
  __syncthreads();
  for (int s = 128; s > 0; s >>= 1) {
    if (t < s) { s_ntg[t] += s_ntg[t+s]; s_tp[t] += s_tp[t+s]; s_ang[t] += s_ang[t+s]; }
    __syncthreads();
  }
  if (t == 0) {
    int npd = 0;
    for (int w = 0; w < NW; ++w) npd += __popc(keepb[b * NW + w]);
    int TP = s_tp[0], NTG = s_ntg[0];
    out[b*3+0] = (float)TP;
    out[b*3+1] = (float)(npd - TP);
    out[b*3+2] = (float)(NTG - TP);
    out[24 + b] = (TP > 0) ? (s_ang[0] / (3.0f * (float)TP)) : 0.0f;
  }
}

// ---------------------------------------------------------------------------
extern "C" void kernel_launch(void* const* d_in, const int* in_sizes, int n_in,
                              void* d_out, int out_size, void* d_ws, size_t ws_size,
                              hipStream_t stream) {
  const float* pred = (const float*)d_in[0];
  const float* targ = (const float*)d_in[1];
  float* out = (float*)d_out;

  // workspace carving (~8 MB total)
  const size_t BN = (size_t)NB * NN;
  float*    W     = (float*)d_ws;
  float*    key   = W;                         // BN
  float*    posu  = key  + BN;                 // BN*3
  float*    Ru    = posu + BN * 3;             // BN*9
  float*    poss  = Ru   + BN * 9;             // BN*3 (sorted, normalized)
  float*    posr  = poss + BN * 3;             // BN*3 (sorted, real)
  float*    Rs    = posr + BN * 3;             // BN*9 (sorted)
  float*    pn2n  = Rs   + BN * 9;             // BN   (|pos|^2 normalized)
  float*    pn2r  = pn2n + BN;                 // BN   (|pos|^2 real)
  float*    tposr = pn2r + BN;                 // BN*3 (target real)
  int*      tact  = (int*)(tposr + BN * 3);    // BN
  float*    tR    = (float*)(tact + BN);       // BN*9
  float*    tn2   = tR + BN * 9;               // BN
  unsigned* keepb = (unsigned*)(tn2 + BN);     // NB*NW
  int*      ftg   = (int*)(keepb + NB * NW);   // BN
  unsigned* adjT  = (unsigned*)(ftg + BN);     // NN*NW (reused per batch)

  k_prep<<<(NB * NN) / 256, 256, 0, stream>>>(pred, targ, key, posu, Ru,
                                              tposr, tact, tR, tn2, ftg);
  k_sortgather<<<NB, 1024, 0, stream>>>(key, posu, Ru, poss, posr, Rs,
                                        pn2n, pn2r, keepb);
  for (int b = 0; b < NB; ++b) {
    k_zero<<<(NN * NW) / 256, 256, 0, stream>>>(adjT);
    k_adj<<<dim3(32, 256), 256, 0, stream>>>(poss + (size_t)b * NN * 3,
                                             pn2n + (size_t)b * NN, adjT);
    k_nms<<<1, 1024, 0, stream>>>(adjT, keepb + b * NW);
  }
  k_match<<<dim3(32, 256, NB), 256, 0, stream>>>(tposr, tn2, tact, posr,
                                                 pn2r, keepb, ftg);
  k_final<<<NB, 256, 0, stream>>>(tact, keepb, ftg, Rs, tR, out);
}